// FSM_40312563040895
// MI455X (gfx1250) — compile-verified
//
#include <hip/hip_runtime.h>

typedef __attribute__((ext_vector_type(2))) float v2f;
typedef __attribute__((ext_vector_type(8))) float v8f;

#define NTIME 2048
#define NCH   256
#define WAVES 16
#define CHUNKS 8            // 8 chunks * 16 steps = 128 timesteps per wave

// final[b,t,c] = 0.5*( s^(t+1)*v0[c] + (1-s)/64 * ema[b,t,c] * v1[c] ) + 0.5*x[b,t,c]
// ema[b,t,c]   = sum_{n<=t} s^(t-n) x[b,n,c]
__global__ __launch_bounds__(512) void FSM_ema_wmma_kernel(
    const float* __restrict__ x,
    const float* __restrict__ sw,
    const float* __restrict__ v0,
    const float* __restrict__ v1,
    float* __restrict__ out)
{
    __shared__ float segEnd[WAVES * 16];
    __shared__ float carries[WAVES * 16];

    const int tid  = threadIdx.x;
    const int wave = tid >> 5;
    const int lane = tid & 31;
    const int half = lane >> 4;     // 0 or 1
    const int n    = lane & 15;     // A-row m / B,C,D column / channel offset
    const int ctile = blockIdx.x;   // 0..15
    const int b     = blockIdx.y;   // 0..31
    const int c     = ctile * 16 + n;
    const int t0    = wave * (CHUNKS * 16);

    // ---- scalar parameters (wave-uniform values; only CONSTANT indices below) ----
    const float w = sw[0];
    const float s = 1.0f / (1.0f + __expf(-w));
    float spow[9];
    spow[0] = 1.0f;
    #pragma unroll
    for (int i = 1; i <= 8; ++i) spow[i] = spow[i - 1] * s;
    const float s8   = spow[8];
    const float s16  = s8 * s8;
    const float s32  = s16 * s16;
    const float s64  = s32 * s32;
    const float s128 = s64 * s64;
    const float K    = (1.0f - s) * (1.0f / 64.0f);   // (1-s)/sqrt(4096)

    // s^(t0+1) for the init-weight term
    float st0p1 = s;
    for (int i = 0; i < wave; ++i) st0p1 *= s128;

    // per-lane s^n via square-and-multiply on bits of n (no dynamic array index)
    float pn = 1.0f;
    {
        float t = s;
        if (n & 1) pn *= t;  t *= t;
        if (n & 2) pn *= t;  t *= t;
        if (n & 4) pn *= t;  t *= t;
        if (n & 8) pn *= t;
    }

    // row-coefficient tables: s^(v+8*half) and s^(v+1+8*half), constant-indexed
    const float sh8 = half ? s8 : 1.0f;
    float initc[8], coefc[8];
    #pragma unroll
    for (int v = 0; v < 8; ++v) {
        initc[v] = spow[v] * sh8;       // s^(mrow)
        coefc[v] = spow[v + 1] * sh8;   // s^(mrow+1)
    }

    // ---- A fragments: L[m,k] = s^(m-k) = s^m * (1/s)^k, lower triangular ----
    const float invs = 1.0f / s;
    float invsp[14];
    invsp[0] = 1.0f;
    #pragma unroll
    for (int i = 1; i <= 13; ++i) invsp[i] = invsp[i - 1] * invs;
    const float invs2h = half ? invsp[2] : 1.0f;

    v2f a[4];
    #pragma unroll
    for (int kb = 0; kb < 4; ++kb) {
        #pragma unroll
        for (int j = 0; j < 2; ++j) {
            const int kgc = 4 * kb + j;          // compile-time part of k
            const int kg  = kgc + 2 * half;      // full k (lane-dependent)
            const float val = pn * invsp[kgc] * invs2h;
            a[kb][j] = (n >= kg) ? val : 0.0f;
        }
    }

    const float* xb = x   + (size_t)b * NTIME * NCH;
    float*       ob = out + (size_t)b * NTIME * NCH;

    // warm HBM->L2/L0 for this wave's segment
    #pragma unroll
    for (int ci = 0; ci < CHUNKS; ++ci)
        __builtin_prefetch(xb + (size_t)(t0 + 16 * ci) * NCH + c, 0, 3);

    // ---- phase 1: wave-local carry-free scan via WMMA ----
    v8f  yloc[CHUNKS];
    float chunkCarry = 0.0f;
    #pragma unroll
    for (int ci = 0; ci < CHUNKS; ++ci) {
        const int tb = t0 + 16 * ci;
        v2f bf[4];
        #pragma unroll
        for (int kb = 0; kb < 4; ++kb) {
            #pragma unroll
            for (int j = 0; j < 2; ++j) {
                const int k = 4 * kb + 2 * half + j;   // time within chunk
                bf[kb][j] = xb[(size_t)(tb + k) * NCH + c];
            }
        }
        v8f acc = {0.0f, 0.0f, 0.0f, 0.0f, 0.0f, 0.0f, 0.0f, 0.0f};
        #pragma unroll
        for (int kb = 0; kb < 4; ++kb)
            acc = __builtin_amdgcn_wmma_f32_16x16x4_f32(
                false, a[kb], false, bf[kb], (short)0, acc, false, false);

        // rank-1 carry from previous chunk: y[mrow] += s^(mrow+1)*carry
        #pragma unroll
        for (int v = 0; v < 8; ++v)
            acc[v] = fmaf(coefc[v], chunkCarry, acc[v]);

        yloc[ci] = acc;
        // value at m=15 for my channel lives in lane (16+n), VGPR 7
        chunkCarry = __int_as_float(
            __builtin_amdgcn_ds_bpermute((16 + n) << 2, __float_as_int(acc[7])));
    }

    // ---- phase 2: combine the 16 wave-segments (serial, trivial) ----
    if (lane < 16) segEnd[wave * 16 + lane] = chunkCarry;
    __syncthreads();
    if (tid < 16) {
        float E = 0.0f;
        #pragma unroll
        for (int wv = 0; wv < WAVES; ++wv) {
            carries[wv * 16 + tid] = E;          // carry INTO segment wv
            E = s128 * E + segEnd[wv * 16 + tid];
        }
    }
    __syncthreads();
    const float waveCarry = carries[wave * 16 + n];

    // ---- phase 3: fixup + blend + store (x re-read hits the 192MB L2) ----
    const float v0h = 0.5f * v0[c];       // 0.5*v0[c]
    const float Kh  = 0.5f * K * v1[c];   // 0.5*(1-s)/64*v1[c]
    float s16p = 1.0f;                    // s^(16*ci)
    #pragma unroll
    for (int ci = 0; ci < CHUNKS; ++ci) {
        const int tb = t0 + 16 * ci;
        const float wc = waveCarry * s16p;      // carry scaled to chunk base
        const float gv = st0p1 * s16p * v0h;    // 0.5*v0[c]*s^(t0+16ci+1)
        #pragma unroll
        for (int v = 0; v < 8; ++v) {
            const int mrow = v + 8 * half;
            const size_t idx = (size_t)(tb + mrow) * NCH + c;
            const float yfull = fmaf(coefc[v], wc, yloc[ci][v]);
            const float xval  = xb[idx];
            ob[idx] = fmaf(initc[v], gv, fmaf(Kh, yfull, 0.5f * xval));
        }
        s16p *= s16;
    }
}

extern "C" void kernel_launch(void* const* d_in, const int* in_sizes, int n_in,
                              void* d_out, int out_size, void* d_ws, size_t ws_size,
                              hipStream_t stream) {
    const float* x  = (const float*)d_in[0];
    const float* sw = (const float*)d_in[1];
    const float* v0 = (const float*)d_in[2];
    const float* v1 = (const float*)d_in[3];
    float* out = (float*)d_out;
    dim3 grid(16, 32);   // (channel tiles, batch)
    dim3 block(512);     // 16 waves of 32
    FSM_ema_wmma_kernel<<<grid, block, 0, stream>>>(x, sw, v0, v1, out);
}